// GAT_30288109371812
// MI455X (gfx1250) — compile-verified
//
#include <hip/hip_runtime.h>
#include <hip/hip_bf16.h>
#include <math.h>

typedef __attribute__((ext_vector_type(2))) float v2f;
typedef __attribute__((ext_vector_type(8))) float v8f;

#define N_NODES 50000
#define F_IN    256
#define HEADS   4
#define C1v     64
#define C2v     32
#define NE      800000
#define ETOT    (NE + N_NODES)
#define NEG_SLOPE 0.2f
#define GEMM_K  256            // K is 256 for both WMMA GEMMs in this model

// ---------------------------------------------------------------------------
// fp32 WMMA GEMM: C[M,Nc] = A[M,K] @ B[K,Nc], K == 256.
// One block (8 waves) covers 8 M-tiles x one 64-col N strip.
// The 256x64 B panel (64KB) is staged to LDS once per block, pre-swizzled into
// WMMA fragment layout, so the inner loop is 1 global_load_b64 (A) +
// 4 conflict-free ds_load_b64 (B frags) per 4 v_wmma_f32_16x16x4_f32.
// A frag: lane holds row lane%16, k-pair (lane<16?0:2); B frag mirrored by
// column; C/D per ISA 8-VGPR striping (M = j + 8*(lane>=16)).
// ---------------------------------------------------------------------------
__global__ __launch_bounds__(256) void wmma_gemm_f32_kernel(
    const float* __restrict__ A, const float* __restrict__ B,
    float* __restrict__ C, int M, int K, int Nc)
{
    __shared__ float ldsB[GEMM_K * 64];          // 64 KB: frag layout
    const int tid   = threadIdx.x;
    const int nbase = blockIdx.y * 64;

    // Stage + swizzle B panel: element (k, col) -> frag (k/4, col/16),
    // lane = (((k>>1)&1)<<4) | (col&15), slot = k&1.
    for (int idx = tid; idx < K * 64; idx += 256) {
        int k     = idx >> 6;
        int col   = idx & 63;
        float v   = B[(size_t)k * Nc + nbase + col];
        int kstep = k >> 2;
        int khalf = (k >> 1) & 1;
        int klo   = k & 1;
        int nt    = col >> 4;
        int ncol  = col & 15;
        int flane = (khalf << 4) | ncol;
        ldsB[(((kstep << 2) | nt) << 6) + (flane << 1) + klo] = v;
    }
    __syncthreads();

    const int lane  = tid & 31;
    const int wave  = tid >> 5;
    const int mtile = blockIdx.x * 8 + wave;
    if (mtile * 16 >= M) return;                 // wave-uniform; after barrier

    const int mrow  = lane & 15;
    const int khalf = (lane >> 4) << 1;          // 0 for lanes 0-15, 2 for 16-31
    const int ncol  = lane & 15;

    v8f acc0 = {}, acc1 = {}, acc2 = {}, acc3 = {};

    const float* Arow = A + (size_t)(mtile * 16 + mrow) * K;
    for (int k0 = 0; k0 < K; k0 += 4) {
        const float2 av = *(const float2*)(Arow + k0 + khalf);
        v2f a; a.x = av.x; a.y = av.y;

        const float* bp = ldsB + (k0 << 6) + (lane << 1);
        const v2f b0 = *(const v2f*)(bp);
        const v2f b1 = *(const v2f*)(bp + 64);
        const v2f b2 = *(const v2f*)(bp + 128);
        const v2f b3 = *(const v2f*)(bp + 192);

        acc0 = __builtin_amdgcn_wmma_f32_16x16x4_f32(false, a, false, b0, (short)0, acc0, false, false);
        acc1 = __builtin_amdgcn_wmma_f32_16x16x4_f32(false, a, false, b1, (short)0, acc1, false, false);
        acc2 = __builtin_amdgcn_wmma_f32_16x16x4_f32(false, a, false, b2, (short)0, acc2, false, false);
        acc3 = __builtin_amdgcn_wmma_f32_16x16x4_f32(false, a, false, b3, (short)0, acc3, false, false);
    }

    const int rbase = mtile * 16 + ((lane >> 4) << 3);
    const int col   = nbase + ncol;
#pragma unroll
    for (int j = 0; j < 8; ++j) {
        float* Crow = C + (size_t)(rbase + j) * Nc + col;
        Crow[0]  = acc0[j];
        Crow[16] = acc1[j];
        Crow[32] = acc2[j];
        Crow[48] = acc3[j];
    }
}

// ---------------------------------------------------------------------------
// Helpers
// ---------------------------------------------------------------------------
__global__ void fill_kernel(float* __restrict__ p, float v, long long n)
{
    long long i = (long long)blockIdx.x * blockDim.x + threadIdx.x;
    if (i < n) p[i] = v;
}

__device__ __forceinline__ void atomicMaxF(float* addr, float v)
{
    // IEEE ordering trick: signed-max for v>=0, unsigned-min for v<0.
    if (v >= 0.0f) atomicMax((int*)addr, __float_as_int(v));
    else           atomicMin((unsigned int*)addr, __float_as_uint(v));
}

// asrc[n,h] = <h[n,h,:], a_src[h,:]> ; adst likewise (one thread per (n,h))
template<int C>
__global__ void node_att_kernel(const float* __restrict__ h,
                                const float* __restrict__ a_src,
                                const float* __restrict__ a_dst,
                                float* __restrict__ asrc,
                                float* __restrict__ adst, int n)
{
    int i = blockIdx.x * blockDim.x + threadIdx.x;
    if (i >= n * HEADS) return;
    int node = i / HEADS, hh = i - node * HEADS;
    const float* hp = h + (size_t)node * HEADS * C + hh * C;
    const float* as = a_src + hh * C;
    const float* ad = a_dst + hh * C;
    float s0 = 0.f, s1 = 0.f;
#pragma unroll
    for (int c = 0; c < C; c += 4) {
        float4 hv = *(const float4*)(hp + c);
        float4 sv = *(const float4*)(as + c);
        float4 dv = *(const float4*)(ad + c);
        s0 += hv.x * sv.x + hv.y * sv.y + hv.z * sv.z + hv.w * sv.w;
        s1 += hv.x * dv.x + hv.y * dv.y + hv.z * dv.z + hv.w * dv.w;
    }
    asrc[i] = s0;
    adst[i] = s1;
}

__device__ __forceinline__ void edge_sd(const int* __restrict__ ei, int i, int& s, int& d)
{
    if (i < NE) { s = ei[i]; d = ei[NE + i]; }
    else        { s = i - NE; d = i - NE; }            // self loops appended
}

__global__ void edge_max_kernel(const int* __restrict__ ei,
                                const float* __restrict__ asrc,
                                const float* __restrict__ adst,
                                float* __restrict__ emax)
{
    int i = blockIdx.x * blockDim.x + threadIdx.x;
    if (i >= ETOT) return;
    int s, d; edge_sd(ei, i, s, d);
#pragma unroll
    for (int h = 0; h < HEADS; ++h) {
        float e = asrc[s * HEADS + h] + adst[d * HEADS + h];
        e = (e > 0.f) ? e : NEG_SLOPE * e;
        atomicMaxF(&emax[d * HEADS + h], e);
    }
}

__global__ void edge_expsum_kernel(const int* __restrict__ ei,
                                   const float* __restrict__ asrc,
                                   const float* __restrict__ adst,
                                   const float* __restrict__ emax,
                                   float* __restrict__ ex,
                                   float* __restrict__ denom)
{
    int i = blockIdx.x * blockDim.x + threadIdx.x;
    if (i >= ETOT) return;
    int s, d; edge_sd(ei, i, s, d);
#pragma unroll
    for (int h = 0; h < HEADS; ++h) {
        float e = asrc[s * HEADS + h] + adst[d * HEADS + h];
        e = (e > 0.f) ? e : NEG_SLOPE * e;
        float v = __expf(e - emax[d * HEADS + h]);
        ex[(size_t)i * HEADS + h] = v;
        atomicAdd(&denom[d * HEADS + h], v);
    }
}

// one wave per edge: gather h[src] (H*C floats, coalesced), scale by per-head
// alpha, atomic-add into out[dst].
template<int C>
__global__ __launch_bounds__(256) void edge_aggregate_kernel(
    const int* __restrict__ ei, const float* __restrict__ hfeat,
    const float* __restrict__ ex, const float* __restrict__ denom,
    float* __restrict__ out)
{
    const int HC = HEADS * C;
    const int F  = HC / 32;                      // features per lane
    const int lane = threadIdx.x & 31;
    int e = blockIdx.x * 8 + (threadIdx.x >> 5); // wave-uniform edge id
    if (e >= ETOT) return;
    int s, d; edge_sd(ei, e, s, d);

    float alpha[HEADS];
#pragma unroll
    for (int h = 0; h < HEADS; ++h)
        alpha[h] = ex[(size_t)e * HEADS + h] / denom[d * HEADS + h];

    const float* hs = hfeat + (size_t)s * HC;
    float*       od = out   + (size_t)d * HC;
#pragma unroll
    for (int j = 0; j < F; ++j) {
        int f = j * 32 + lane;
        int h = f / C;
        atomicAdd(&od[f], hs[f] * alpha[h]);
    }
}

__global__ void bias_relu_kernel(float* __restrict__ x, const float* __restrict__ b,
                                 int fmask, long long n)
{
    long long i = (long long)blockIdx.x * blockDim.x + threadIdx.x;
    if (i >= n) return;
    float v = x[i] + b[(int)i & fmask];          // F is a power of two
    x[i] = v > 0.f ? v : 0.f;
}

__global__ void deg_kernel(const int* __restrict__ ei, float* __restrict__ deg)
{
    int i = blockIdx.x * blockDim.x + threadIdx.x;
    if (i >= ETOT) return;
    int s, d; edge_sd(ei, i, s, d);
    (void)s;
    atomicAdd(&deg[d], 1.0f);
}

__global__ void dinv_kernel(float* __restrict__ deg, int n)
{
    int i = blockIdx.x * blockDim.x + threadIdx.x;
    if (i >= n) return;
    deg[i] = rsqrtf(fmaxf(deg[i], 1e-12f));
}

// [N,128] @ [128,2] — tiny N-dim, plain dot products (memory bound anyway)
__global__ void gcn_lin_kernel(const float* __restrict__ x, const float* __restrict__ W,
                               float* __restrict__ h3, int n)
{
    int node = blockIdx.x * blockDim.x + threadIdx.x;
    if (node >= n) return;
    const float* xp = x + (size_t)node * 128;
    float a0 = 0.f, a1 = 0.f;
#pragma unroll
    for (int k = 0; k < 128; k += 4) {
        float4 v = *(const float4*)(xp + k);
        a0 += v.x * W[(k + 0) * 2] + v.y * W[(k + 1) * 2] + v.z * W[(k + 2) * 2] + v.w * W[(k + 3) * 2];
        a1 += v.x * W[(k + 0) * 2 + 1] + v.y * W[(k + 1) * 2 + 1] + v.z * W[(k + 2) * 2 + 1] + v.w * W[(k + 3) * 2 + 1];
    }
    h3[node * 2] = a0;
    h3[node * 2 + 1] = a1;
}

__global__ void gcn_agg_kernel(const int* __restrict__ ei, const float* __restrict__ h3,
                               const float* __restrict__ dinv, float* __restrict__ out)
{
    int i = blockIdx.x * blockDim.x + threadIdx.x;
    if (i >= ETOT) return;
    int s, d; edge_sd(ei, i, s, d);
    float norm = dinv[s] * dinv[d];
    atomicAdd(&out[2 * d],     h3[2 * s] * norm);
    atomicAdd(&out[2 * d + 1], h3[2 * s + 1] * norm);
}

__global__ void logsoftmax_kernel(float* __restrict__ out, const float* __restrict__ b3, int n)
{
    int i = blockIdx.x * blockDim.x + threadIdx.x;
    if (i >= n) return;
    float v0 = out[2 * i] + b3[0];
    float v1 = out[2 * i + 1] + b3[1];
    float m = fmaxf(v0, v1);
    float lse = m + logf(__expf(v0 - m) + __expf(v1 - m));
    out[2 * i] = v0 - lse;
    out[2 * i + 1] = v1 - lse;
}

// ---------------------------------------------------------------------------
extern "C" void kernel_launch(void* const* d_in, const int* in_sizes, int n_in,
                              void* d_out, int out_size, void* d_ws, size_t ws_size,
                              hipStream_t stream)
{
    const float* x        = (const float*)d_in[0];
    const int*   ei       = (const int*)d_in[1];
    const float* W1       = (const float*)d_in[2];
    const float* att_src1 = (const float*)d_in[3];
    const float* att_dst1 = (const float*)d_in[4];
    const float* b1       = (const float*)d_in[5];
    const float* W2       = (const float*)d_in[6];
    const float* att_src2 = (const float*)d_in[7];
    const float* att_dst2 = (const float*)d_in[8];
    const float* b2       = (const float*)d_in[9];
    const float* W3       = (const float*)d_in[10];
    const float* b3       = (const float*)d_in[11];
    float* out = (float*)d_out;

    // workspace layout (floats)
    float* ws = (float*)d_ws;
    size_t o = 0;
    float* bufA  = ws + o; o += (size_t)N_NODES * 256;   // h1 -> later h2
    float* bufB  = ws + o; o += (size_t)N_NODES * 256;   // out1 -> later out2
    float* ex    = ws + o; o += (size_t)ETOT * HEADS;
    float* asrc  = ws + o; o += (size_t)N_NODES * HEADS;
    float* adst  = ws + o; o += (size_t)N_NODES * HEADS;
    float* emax  = ws + o; o += (size_t)N_NODES * HEADS;
    float* denom = ws + o; o += (size_t)N_NODES * HEADS;
    float* h3    = ws + o; o += (size_t)N_NODES * 2;
    float* deg   = ws + o; o += (size_t)N_NODES;          // becomes dinv

    const long long NH = (long long)N_NODES * HEADS;
    const int TB = 256;
    const int egrid  = (ETOT + TB - 1) / TB;              // thread-per-edge kernels
    const int ewgrid = (ETOT + 7) / 8;                    // wave-per-edge kernels
    const int nhgrid = (int)((NH + TB - 1) / TB);
    const int ngrid  = (N_NODES + TB - 1) / TB;
    const int mblocks = (N_NODES / 16 + 7) / 8;           // 3125 M-tiles / 8 waves

    // ---------------- GAT layer 1 ----------------
    fill_kernel<<<nhgrid, TB, 0, stream>>>(emax, -INFINITY, NH);
    fill_kernel<<<nhgrid, TB, 0, stream>>>(denom, 0.f, NH);
    fill_kernel<<<(int)(((long long)N_NODES * 256 + TB - 1) / TB), TB, 0, stream>>>(
        bufB, 0.f, (long long)N_NODES * 256);

    wmma_gemm_f32_kernel<<<dim3(mblocks, 4), TB, 0, stream>>>(x, W1, bufA, N_NODES, 256, 256);
    node_att_kernel<C1v><<<nhgrid, TB, 0, stream>>>(bufA, att_src1, att_dst1, asrc, adst, N_NODES);
    edge_max_kernel<<<egrid, TB, 0, stream>>>(ei, asrc, adst, emax);
    edge_expsum_kernel<<<egrid, TB, 0, stream>>>(ei, asrc, adst, emax, ex, denom);
    edge_aggregate_kernel<C1v><<<ewgrid, TB, 0, stream>>>(ei, bufA, ex, denom, bufB);
    bias_relu_kernel<<<(int)(((long long)N_NODES * 256 + TB - 1) / TB), TB, 0, stream>>>(
        bufB, b1, 255, (long long)N_NODES * 256);

    // ---------------- GAT layer 2 ----------------
    wmma_gemm_f32_kernel<<<dim3(mblocks, 2), TB, 0, stream>>>(bufB, W2, bufA, N_NODES, 256, 128);
    fill_kernel<<<nhgrid, TB, 0, stream>>>(emax, -INFINITY, NH);
    fill_kernel<<<nhgrid, TB, 0, stream>>>(denom, 0.f, NH);
    fill_kernel<<<(int)(((long long)N_NODES * 128 + TB - 1) / TB), TB, 0, stream>>>(
        bufB, 0.f, (long long)N_NODES * 128);
    node_att_kernel<C2v><<<nhgrid, TB, 0, stream>>>(bufA, att_src2, att_dst2, asrc, adst, N_NODES);
    edge_max_kernel<<<egrid, TB, 0, stream>>>(ei, asrc, adst, emax);
    edge_expsum_kernel<<<egrid, TB, 0, stream>>>(ei, asrc, adst, emax, ex, denom);
    edge_aggregate_kernel<C2v><<<ewgrid, TB, 0, stream>>>(ei, bufA, ex, denom, bufB);
    bias_relu_kernel<<<(int)(((long long)N_NODES * 128 + TB - 1) / TB), TB, 0, stream>>>(
        bufB, b2, 127, (long long)N_NODES * 128);

    // ---------------- GCN layer + log_softmax ----------------
    fill_kernel<<<ngrid, TB, 0, stream>>>(deg, 0.f, N_NODES);
    deg_kernel<<<egrid, TB, 0, stream>>>(ei, deg);
    dinv_kernel<<<ngrid, TB, 0, stream>>>(deg, N_NODES);
    gcn_lin_kernel<<<ngrid, TB, 0, stream>>>(bufB, W3, h3, N_NODES);
    fill_kernel<<<(int)(((long long)N_NODES * 2 + TB - 1) / TB), TB, 0, stream>>>(
        out, 0.f, (long long)N_NODES * 2);
    gcn_agg_kernel<<<egrid, TB, 0, stream>>>(ei, h3, deg, out);
    logsoftmax_kernel<<<ngrid, TB, 0, stream>>>(out, b3, N_NODES);
}